// DotProductAtten_85048942395855
// MI455X (gfx1250) — compile-verified
//
#include <hip/hip_runtime.h>
#include <math.h>

typedef float v2f __attribute__((ext_vector_type(2)));
typedef float v8f __attribute__((ext_vector_type(8)));
typedef int   v4i __attribute__((ext_vector_type(4)));

#define D_      64
#define QT      16               // query rows per wave
#define KT      64               // keys per LDS tile (double-buffered)
#define WAVES   8                // 256 threads
#define KSTRIDE 68               // padded LDS row stride (floats): 16B rows, conflict-free reads
#define TILE_F  (KT * KSTRIDE)   // floats per tile buffer
#define PSTRIDE 18               // padded P scratch stride
#define NEG_BIG -1.0e6f

#if __has_builtin(__builtin_amdgcn_global_load_async_to_lds_b128)
#define USE_ASYNC 1
#else
#define USE_ASYNC 0
#endif

#if USE_ASYNC
// builtin expects (int4 AS1*, int4 AS3*, imm offset, imm cpol)
#define GPTR(p) ((__attribute__((address_space(1))) v4i*)(p))
#define LPTR(p) ((__attribute__((address_space(3))) v4i*)(p))
#if __has_builtin(__builtin_amdgcn_s_wait_asynccnt)
#define ASYNC_WAIT(n) __builtin_amdgcn_s_wait_asynccnt(n)
#else
#define ASYNC_WAIT(n) asm volatile("s_wait_asynccnt %0" ::"i"(n) : "memory")
#endif
#else
#define ASYNC_WAIT(n) do {} while (0)
#endif

__global__ __launch_bounds__(256)
void fa_f32_wmma_kernel(const float* __restrict__ q,
                        const float* __restrict__ k,
                        const float* __restrict__ v,
                        const int*   __restrict__ vlen,
                        float* __restrict__ out,
                        int S)
{
    __shared__ float sK[2 * TILE_F];             // K tiles, row-major [key][feat]
    __shared__ float sV[2 * TILE_F];             // V tiles, row-major [key][feat]
    __shared__ float sP[WAVES][QT * PSTRIDE];    // per-wave P subtile scratch

    const int tid   = threadIdx.x;
    const int wid   = tid >> 5;
    const int lane  = tid & 31;
    const int lo    = lane & 15;          // column / row-in-half index
    const int hb    = (lane >> 4) << 1;   // K-pair offset for A/B operands (0 or 2)
    const int rhalf = (lane >> 4) << 3;   // row offset for C/D layout (0 or 8)

    const int blocksPerBatch = S / (WAVES * QT);
    const int b      = blockIdx.x / blocksPerBatch;
    const int qtile  = (blockIdx.x % blocksPerBatch) * (WAVES * QT) + wid * QT;

    const int valid = vlen[b];
    const int nkt   = (valid + KT - 1) / KT;   // tiles fully past valid_len skipped:
                                               // exp(-1e6 - m) == 0 in fp32, identical result
    const size_t bbase = (size_t)b * S * D_;

    const float4* gK = (const float4*)(k + bbase);
    const float4* gV = (const float4*)(v + bbase);

    // ---- stage one 64-key K/V tile into LDS buffer `buf` (8 x async b128 per thread) ----
    auto stage = [&](int kt, int buf) {
        const int base4 = kt * (KT * D_ / 4);    // float4 index of tile start
        float* sKb = &sK[buf * TILE_F];
        float* sVb = &sV[buf * TILE_F];
#pragma unroll
        for (int i = 0; i < 4; ++i) {
            const int idx4 = tid + i * 256;      // 1024 float4 per tile
            const int row  = idx4 >> 4;
            const int c0   = (idx4 & 15) << 2;
            float* lk = &sKb[row * KSTRIDE + c0];   // (row*68+c0)*4 is 16B aligned
            float* lv = &sVb[row * KSTRIDE + c0];
#if USE_ASYNC
            __builtin_amdgcn_global_load_async_to_lds_b128(
                GPTR(gK + base4 + idx4), LPTR(lk), 0, 0);
            __builtin_amdgcn_global_load_async_to_lds_b128(
                GPTR(gV + base4 + idx4), LPTR(lv), 0, 0);
#else
            *(float4*)lk = gK[base4 + idx4];
            *(float4*)lv = gV[base4 + idx4];
#endif
        }
    };

    // ---- Q tile as WMMA A-operands (scaled by 1/sqrt(D) = 0.125) ----
    v2f aq[16];
    {
        const float* qrow = q + bbase + (size_t)(qtile + lo) * D_;
#pragma unroll
        for (int s = 0; s < 16; ++s) {
            v2f t = *(const v2f*)(qrow + s * 4 + hb);
            t[0] *= 0.125f; t[1] *= 0.125f;
            aq[s] = t;
        }
    }

    v8f o[4];
#pragma unroll
    for (int n = 0; n < 4; ++n) o[n] = {};
    float m[8], l[8];
#pragma unroll
    for (int r = 0; r < 8; ++r) { m[r] = -INFINITY; l[r] = 0.0f; }

    stage(0, 0);

    for (int kt = 0; kt < nkt; ++kt) {
        const int cur = kt & 1;
        if (kt + 1 < nkt) {
            stage(kt + 1, cur ^ 1);   // issue next tile's async copies first...
            ASYNC_WAIT(8);            // ...then wait: in-order => current tile landed
        } else {
            ASYNC_WAIT(0);
        }
        __syncthreads();

        const float* sKb = &sK[cur * TILE_F];
        const float* sVb = &sV[cur * TILE_F];
        const int kstart = kt * KT;

        // ---- S = (Q/8) . K^T : 16x64 scores, 4 N-subtiles x 16 K-steps ----
        v8f sc[4];
#pragma unroll
        for (int j = 0; j < 4; ++j) {
            v8f acc = {};
#pragma unroll
            for (int s = 0; s < 16; ++s) {
                v2f bk = *(const v2f*)(&sKb[(j * 16 + lo) * KSTRIDE + s * 4 + hb]);
                acc = __builtin_amdgcn_wmma_f32_16x16x4_f32(
                        false, aq[s], false, bk, (short)0, acc, false, false);
            }
            if (kstart + j * 16 + lo >= valid) {   // key-mask, per-lane uniform in C/D layout
#pragma unroll
                for (int r = 0; r < 8; ++r) acc[r] = NEG_BIG;
            }
            sc[j] = acc;
        }

        // ---- online softmax (row stats live per-16-lane-half in C/D layout) ----
        float mn[8], scl[8];
#pragma unroll
        for (int r = 0; r < 8; ++r) {
            float t = fmaxf(fmaxf(sc[0][r], sc[1][r]), fmaxf(sc[2][r], sc[3][r]));
#pragma unroll
            for (int msk = 8; msk >= 1; msk >>= 1)
                t = fmaxf(t, __shfl_xor(t, msk, 32));
            mn[r]  = fmaxf(m[r], t);
            scl[r] = __expf(m[r] - mn[r]);
            m[r]   = mn[r];
        }
#pragma unroll
        for (int j = 0; j < 4; ++j)
#pragma unroll
            for (int r = 0; r < 8; ++r)
                sc[j][r] = __expf(sc[j][r] - mn[r]);
#pragma unroll
        for (int r = 0; r < 8; ++r) {
            float t = (sc[0][r] + sc[1][r]) + (sc[2][r] + sc[3][r]);
#pragma unroll
            for (int msk = 8; msk >= 1; msk >>= 1)
                t += __shfl_xor(t, msk, 32);
            l[r] = l[r] * scl[r] + t;
        }
#pragma unroll
        for (int n = 0; n < 4; ++n)
#pragma unroll
            for (int r = 0; r < 8; ++r)
                o[n][r] *= scl[r];

        // ---- O += P . V : per 16-key subtile, C/D->A relayout via LDS scratch ----
        float* pw = sP[wid];
#pragma unroll
        for (int j = 0; j < 4; ++j) {
#pragma unroll
            for (int r = 0; r < 8; ++r)
                pw[(r + rhalf) * PSTRIDE + lo] = sc[j][r];
            // same-wave LDS ops are in-order; compiler inserts s_wait_dscnt for the RAW
#pragma unroll
            for (int n = 0; n < 4; ++n) {
#pragma unroll
                for (int s = 0; s < 4; ++s) {
                    const int kidx = j * 16 + s * 4 + hb;
                    v2f ap = *(const v2f*)(&pw[lo * PSTRIDE + s * 4 + hb]);
                    v2f bv;                       // B[k][n], B[k+1][n] from row-major V:
                    bv[0] = sVb[(kidx + 0) * KSTRIDE + n * 16 + lo];   // -> ds_load_2addr_b32
                    bv[1] = sVb[(kidx + 1) * KSTRIDE + n * 16 + lo];
                    o[n] = __builtin_amdgcn_wmma_f32_16x16x4_f32(
                             false, ap, false, bv, (short)0, o[n], false, false);
                }
            }
        }
        __syncthreads();
    }

    // ---- normalize and write out ----
    float inv[8];
#pragma unroll
    for (int r = 0; r < 8; ++r) inv[r] = 1.0f / l[r];
#pragma unroll
    for (int n = 0; n < 4; ++n)
#pragma unroll
        for (int r = 0; r < 8; ++r) {
            const int row = qtile + r + rhalf;
            out[bbase + (size_t)row * D_ + n * 16 + lo] = o[n][r] * inv[r];
        }
}

extern "C" void kernel_launch(void* const* d_in, const int* in_sizes, int n_in,
                              void* d_out, int out_size, void* d_ws, size_t ws_size,
                              hipStream_t stream) {
    const float* q  = (const float*)d_in[0];
    const float* k  = (const float*)d_in[1];
    const float* v  = (const float*)d_in[2];
    const int* vlen = (const int*)d_in[3];
    float* out = (float*)d_out;

    const int B = in_sizes[3];                 // [B] valid_lens
    const int S = in_sizes[0] / (B * D_);      // q is [B,S,D]
    const int blocksPerBatch = S / (WAVES * QT);
    const int grid = B * blocksPerBatch;

    fa_f32_wmma_kernel<<<dim3(grid), dim3(256), 0, stream>>>(q, k, v, vlen, out, S);
}